// QuantizedRPE_58815282151821
// MI455X (gfx1250) — compile-verified
//
#include <hip/hip_runtime.h>
#include <math.h>

// Problem constants (from reference): coords [4,1024,2] f32, rpe_table [64,16] f32,
// output [4,16,1024,1024] f32.
#define BB   4
#define TT   1024
#define HH   16
#define BINS 32
#define HALF 16
#define PAD  20   // padded LDS row stride in floats: 80B = 16B-aligned, spreads banks

typedef float v4f __attribute__((ext_vector_type(4)));

__device__ __forceinline__ v4f lds_load4(const float* p) { return *(const v4f*)p; }

// ---------------------------------------------------------------------------
// Kernel 1: eta_range = max(max_b(max(eta_b) - min(eta_b)), 1e-6)  -> ws[0]
// ---------------------------------------------------------------------------
__global__ __launch_bounds__(256) void rpe_range_kernel(const float* __restrict__ coords,
                                                        float* __restrict__ ws) {
  __shared__ float smax[256];
  __shared__ float smin[256];
  const int t = threadIdx.x;
  float range = 0.0f;
  for (int b = 0; b < BB; ++b) {
    float mx = -3.402823466e38f, mn = 3.402823466e38f;
    for (int k = t; k < TT; k += 256) {
      float e = coords[(b * TT + k) * 2];  // eta = coords[...,0]
      mx = fmaxf(mx, e);
      mn = fminf(mn, e);
    }
    smax[t] = mx;
    smin[t] = mn;
    __syncthreads();
    for (int s = 128; s > 0; s >>= 1) {
      if (t < s) {
        smax[t] = fmaxf(smax[t], smax[t + s]);
        smin[t] = fminf(smin[t], smin[t + s]);
      }
      __syncthreads();
    }
    if (t == 0) range = fmaxf(range, smax[0] - smin[0]);
    __syncthreads();
  }
  if (t == 0) ws[0] = fmaxf(range, 1e-6f);
}

// ---------------------------------------------------------------------------
// Kernel 2: one block per (b, i) row; 256 threads; thread owns 4 consecutive j.
// Index math once per (i,j), amortized over 16 heads. NT b128 stores.
// ---------------------------------------------------------------------------
__global__ __launch_bounds__(256) void rpe_bias_kernel(const float* __restrict__ coords,
                                                       const float* __restrict__ rpe_table,
                                                       const float* __restrict__ ws,
                                                       float* __restrict__ out) {
  __shared__ __align__(16) float tab[2 * BINS * PAD];  // 64 rows x 20 floats = 5120 B

  const int t = threadIdx.x;
  const int b = blockIdx.x >> 10;
  const int i = blockIdx.x & (TT - 1);

  // Stage 64x16 table into padded LDS via CDNA5 async copy (ASYNCcnt path).
  // Element e of the packed table -> padded row layout row*PAD + col.
  for (int e = t; e < 2 * BINS * HH; e += 256) {
    const int row = e >> 4;
    const int col = e & 15;
    unsigned lds_off = (unsigned)(size_t)(&tab[row * PAD + col]);  // LDS byte offset
    unsigned g_off = (unsigned)(e * 4);                             // global byte offset
    asm volatile("global_load_async_to_lds_b32 %0, %1, %2"
                 :
                 : "v"(lds_off), "v"(g_off), "s"(rpe_table)
                 : "memory");
  }
  asm volatile("s_wait_asynccnt 0" ::: "memory");
  __syncthreads();

  const float eta_range = ws[0];  // uniform -> s_load
  const float eta_i = coords[(b * TT + i) * 2 + 0];
  const float phi_i = coords[(b * TT + i) * 2 + 1];

  const float PI_F   = 3.14159265358979323846f;
  const float TWO_PI = 6.28318530717958647692f;

  // Load 4 consecutive (eta,phi) pairs: j0 = 4*t .. 4*t+3
  const v4f* crow = (const v4f*)(coords + (size_t)b * TT * 2);
  v4f p0 = crow[t * 2 + 0];  // eta,phi for j0, j0+1
  v4f p1 = crow[t * 2 + 1];  // eta,phi for j0+2, j0+3

  float ej[4] = {p0.x, p0.z, p1.x, p1.z};
  float pj[4] = {p0.y, p0.w, p1.y, p1.w};

  int erow[4], prow[4];
#pragma unroll
  for (int k = 0; k < 4; ++k) {
    float rel_eta = eta_i - ej[k];
    // rel_phi = mod(dphi + pi, 2pi) - pi   (floor-mod, matches jnp.mod)
    float w = phi_i - pj[k] + PI_F;
    w = w - TWO_PI * floorf(w / TWO_PI);
    float rel_phi = w - PI_F;
    // trunc-toward-zero cast, exactly as reference order of ops (div then *16)
    int eb = (int)(rel_eta / eta_range * (float)HALF);
    int pb = (int)(rel_phi / PI_F * (float)HALF);
    eb = min(max(eb, -HALF), HALF - 1) + HALF;          // [0,32)
    pb = min(max(pb, -HALF), HALF - 1) + HALF + BINS;   // [32,64)
    erow[k] = eb * PAD;
    prow[k] = pb * PAD;
  }

  v4f* out4 = (v4f*)out;
#pragma unroll
  for (int h4 = 0; h4 < 4; ++h4) {
    union { v4f v; float f[4]; } s[4];
#pragma unroll
    for (int k = 0; k < 4; ++k) {
      v4f a = lds_load4(&tab[erow[k] + h4 * 4]);  // ds_load_b128
      v4f c = lds_load4(&tab[prow[k] + h4 * 4]);  // ds_load_b128
      s[k].v = a + c;
    }
#pragma unroll
    for (int hh = 0; hh < 4; ++hh) {
      const int h = h4 * 4 + hh;
      // out[b][h][i][j], float4 index: ((b*H+h)*T + i) * (T/4) + t
      size_t vidx = ((size_t)((b * HH + h) * TT + i) << 8) + (size_t)t;
      v4f stv = {s[0].f[hh], s[1].f[hh], s[2].f[hh], s[3].f[hh]};
      __builtin_nontemporal_store(stv, &out4[vidx]);  // streaming 256MB, bypass-ish L2 hint
    }
  }
}

// ---------------------------------------------------------------------------
extern "C" void kernel_launch(void* const* d_in, const int* in_sizes, int n_in,
                              void* d_out, int out_size, void* d_ws, size_t ws_size,
                              hipStream_t stream) {
  const float* coords    = (const float*)d_in[0];   // [4,1024,2]
  const float* rpe_table = (const float*)d_in[1];   // [64,16]
  float* ws  = (float*)d_ws;
  float* out = (float*)d_out;                       // [4,16,1024,1024]

  hipLaunchKernelGGL(rpe_range_kernel, dim3(1), dim3(256), 0, stream, coords, ws);
  hipLaunchKernelGGL(rpe_bias_kernel, dim3(BB * TT), dim3(256), 0, stream,
                     coords, rpe_table, ws, out);
}